// FCOSWithTRTNMS_73538430042612
// MI455X (gfx1250) — compile-verified
//
#include <hip/hip_runtime.h>
#include <math.h>

// FCOS postprocess + NMS for MI455X (gfx1250, wave32).
//
// Pipeline (all on `stream`, scratch in d_ws):
//   1) fcos_decode_score : 1 wave per row; argmax over 91 logits (monotone
//      sigmoid -> argmax on raw logits), 1 sigmoid pair per row, float4 box
//      decode. Pure bandwidth: ~35 MB => ~1.5us at 23.3 TB/s.
//   2) histogram + threshold + compaction -> deterministic top-<=256 set.
//   3) nms_select (single block, LDS-resident): bitonic sort by (score desc,
//      idx asc), pairwise suppression matrix where area_i+area_j per 16x16
//      tile is a rank-2 outer product done with v_wmma_f32_16x16x4_f32;
//      intersection via VALU from LDS; bits packed via __ballot; 256-step
//      serial bitmask scan selects 100 outputs.

#define NCLS   91
#define KMAX   256
#define NBINS  4096
#define IOU_T  0.6f
#define MAXOUT 100

typedef __attribute__((ext_vector_type(2))) float v2f;
typedef __attribute__((ext_vector_type(8))) float v8f;

__device__ __forceinline__ float sigmoidf_(float x) {
    return 1.0f / (1.0f + __expf(-x));
}

// ---------------------------------------------------------------- phase 1
__global__ void fcos_decode_score(const float* __restrict__ logits,
                                  const float* __restrict__ reg,
                                  const float* __restrict__ ctrness,
                                  const float* __restrict__ anchors,
                                  float* __restrict__ boxes,
                                  float* __restrict__ scores,
                                  int* __restrict__ labels, int n) {
    const int wavesPerBlock = blockDim.x >> 5;
    const int row  = blockIdx.x * wavesPerBlock + (threadIdx.x >> 5);
    const int lane = threadIdx.x & 31;
    if (row >= n) return;

    const float* rp = logits + (size_t)row * NCLS;
    float best = -1e30f;
    int   bidx = 0x7fffffff;
#pragma unroll
    for (int c0 = 0; c0 < NCLS; c0 += 32) {
        int c = c0 + lane;
        if (c < NCLS) {
            float v = rp[c];
            if (v > best) { best = v; bidx = c; }   // lowest index on tie
        }
    }
    // wave32 max+argmax reduction (lowest class index wins ties)
    for (int off = 16; off > 0; off >>= 1) {
        float ov = __shfl_xor(best, off, 32);
        int   oi = __shfl_xor(bidx, off, 32);
        if (ov > best || (ov == best && oi < bidx)) { best = ov; bidx = oi; }
    }
    if (lane == 0) {
        float s = sqrtf(sigmoidf_(best) * sigmoidf_(ctrness[row]));
        scores[row] = s;
        labels[row] = bidx;
        float4 a = ((const float4*)anchors)[row];
        float4 r = ((const float4*)reg)[row];
        float cx = 0.5f * (a.x + a.z), cy = 0.5f * (a.y + a.w);
        float w  = a.z - a.x,          h  = a.w - a.y;
        float4 b;
        b.x = cx - r.x * w; b.y = cy - r.y * h;
        b.z = cx + r.z * w; b.w = cy + r.w * h;
        ((float4*)boxes)[row] = b;
    }
}

// ---------------------------------------------------------------- phase 2
__global__ void nms_init(int* __restrict__ hist, int* __restrict__ counter,
                         int* __restrict__ thrbin) {
    for (int i = threadIdx.x; i < NBINS; i += blockDim.x) hist[i] = 0;
    if (threadIdx.x == 0) { *counter = 0; *thrbin = NBINS - 1; }
}

__device__ __forceinline__ int score_bin(float s) {
    int b = (int)(s * (float)NBINS);
    return b < 0 ? 0 : (b > NBINS - 1 ? NBINS - 1 : b);
}

__global__ void score_hist(const float* __restrict__ scores,
                           int* __restrict__ hist, int n) {
    int i = blockIdx.x * blockDim.x + threadIdx.x;
    if (i < n) atomicAdd(&hist[score_bin(scores[i])], 1);
}

__global__ void pick_threshold(const int* __restrict__ hist,
                               int* __restrict__ thrbin) {
    if (threadIdx.x == 0) {
        int acc = 0, t = NBINS - 1;
        for (int b = NBINS - 1; b >= 0; b--) {
            if (acc + hist[b] > KMAX) break;
            acc += hist[b];
            t = b;
        }
        *thrbin = t;  // suffix count over [t, NBINS) is <= KMAX
    }
}

__global__ void compact_cands(const float* __restrict__ scores,
                              const int* __restrict__ thrbin,
                              int* __restrict__ counter,
                              int* __restrict__ cand, int n) {
    int i = blockIdx.x * blockDim.x + threadIdx.x;
    if (i < n) {
        if (score_bin(scores[i]) >= *thrbin) {
            int p = atomicAdd(counter, 1);
            if (p < KMAX) cand[p] = i;
        }
    }
}

// ---------------------------------------------------------------- phase 3
__global__ void __launch_bounds__(256)
nms_select(const float* __restrict__ boxes, const float* __restrict__ scores,
           const int* __restrict__ labels, const int* __restrict__ cand,
           const int* __restrict__ counter, float* __restrict__ out) {
    __shared__ float skey[KMAX];
    __shared__ int   sidx[KMAX];
    __shared__ float bx1[KMAX], by1[KMAX], bx2[KMAX], by2[KMAX], sarea[KMAX];
    __shared__ unsigned smask[KMAX * (KMAX / 32)];   // suppression matrix bits
    __shared__ unsigned suppVec[KMAX / 32];
    __shared__ int keepList[MAXOUT];

    const int tid = threadIdx.x;
    int count = *counter;
    if (count > KMAX) count = KMAX;

    if (tid < count) {
        int oi   = cand[tid];
        skey[tid] = scores[oi];
        sidx[tid] = oi;
    } else {
        skey[tid] = -1e30f;
        sidx[tid] = 0;
    }
    __syncthreads();

    // Bitonic sort, order: score desc, original index asc (deterministic).
    for (int k = 2; k <= KMAX; k <<= 1) {
        for (int j = k >> 1; j > 0; j >>= 1) {
            __syncthreads();
            int ixj = tid ^ j;
            if (ixj > tid) {
                float s1 = skey[tid], s2 = skey[ixj];
                int   i1 = sidx[tid], i2 = sidx[ixj];
                bool bef_ixj = (s2 > s1) || (s2 == s1 && i2 < i1);
                bool bef_tid = (s1 > s2) || (s1 == s2 && i1 < i2);
                bool doswap  = ((tid & k) == 0) ? bef_ixj : bef_tid;
                if (doswap) {
                    skey[tid] = s2; skey[ixj] = s1;
                    sidx[tid] = i2; sidx[ixj] = i1;
                }
            }
        }
    }
    __syncthreads();

    // Gather boxes of sorted candidates into LDS; pad with degenerate boxes.
    if (skey[tid] > -1e29f) {
        float4 b = ((const float4*)boxes)[sidx[tid]];
        bx1[tid] = b.x; by1[tid] = b.y; bx2[tid] = b.z; by2[tid] = b.w;
        sarea[tid] = fmaxf(b.z - b.x, 0.0f) * fmaxf(b.w - b.y, 0.0f);
    } else {
        bx1[tid] = 0.0f; by1[tid] = 0.0f; bx2[tid] = 0.0f; by2[tid] = 0.0f;
        sarea[tid] = 0.0f;  // zero area -> inter=0 -> never suppresses
    }
    for (int m = tid; m < KMAX * (KMAX / 32); m += blockDim.x) smask[m] = 0u;
    if (tid < KMAX / 32) suppVec[tid] = 0u;
    __syncthreads();

    // Pairwise suppression matrix over upper-triangular 16x16 tiles.
    // area_i + area_j computed as a rank-2 outer product on the matrix pipe:
    //   A(16x4) = [area_i, 1, 0, 0] rows, B(4x16) = [1; area_j; 0; 0] cols
    //   => D[M][N] = area_{i0+M} + area_{j0+N}  (v_wmma_f32_16x16x4_f32)
    const int waveId = tid >> 5, lane = tid & 31;
    const int NTI  = KMAX / 16;
    const int NTRI = NTI * (NTI + 1) / 2;
    for (int t = waveId; t < NTRI; t += (int)(blockDim.x >> 5)) {
        int ti = 0, rem = t;
        while (rem >= NTI - ti) { rem -= NTI - ti; ti++; }
        int tj = ti + rem;

        bool lo16 = lane < 16;
        v2f av, bv;
        // A 16x4 f32 layout: lanes 0-15 hold K=0,1 ; lanes 16-31 hold K=2,3.
        av[0] = lo16 ? sarea[ti * 16 + lane] : 0.0f;  // K=0 col = area_i
        av[1] = lo16 ? 1.0f : 0.0f;                   // K=1 col = ones
        bv[0] = lo16 ? 1.0f : 0.0f;                   // K=0 row = ones
        bv[1] = lo16 ? sarea[tj * 16 + lane] : 0.0f;  // K=1 row = area_j
        v8f cz = {};
        v8f d = __builtin_amdgcn_wmma_f32_16x16x4_f32(
            false, av, false, bv, (short)0, cz, false, false);

        int   jn  = tj * 16 + (lane & 15);            // column index N
        float jx1 = bx1[jn], jy1 = by1[jn], jx2 = bx2[jn], jy2 = by2[jn];
        int   rbase = ti * 16 + ((lane >= 16) ? 8 : 0);
        unsigned sh = (unsigned)((tj & 1) << 4);
#pragma unroll
        for (int r = 0; r < 8; r++) {
            int  im  = rbase + r;                     // row index M
            bool sup = false;
            if (jn > im) {
                float ix1 = fmaxf(bx1[im], jx1);
                float iy1 = fmaxf(by1[im], jy1);
                float ix2 = fminf(bx2[im], jx2);
                float iy2 = fminf(by2[im], jy2);
                float inter = fmaxf(ix2 - ix1, 0.0f) * fmaxf(iy2 - iy1, 0.0f);
                float uni   = d[r] - inter;           // a_i + a_j - inter (WMMA)
                sup = inter > IOU_T * uni;            // iou > 0.6
            }
            unsigned long long bal = __ballot(sup);
            if (lane == 0) {
                unsigned lobits = (unsigned)(bal & 0xFFFFull);
                if (lobits)
                    atomicOr(&smask[(ti * 16 + r) * (KMAX / 32) + (tj >> 1)],
                             lobits << sh);
            }
            if (lane == 16) {
                unsigned hibits = (unsigned)((bal >> 16) & 0xFFFFull);
                if (hibits)
                    atomicOr(&smask[(ti * 16 + r + 8) * (KMAX / 32) + (tj >> 1)],
                             hibits << sh);
            }
        }
    }
    __syncthreads();

    // Serial greedy scan over sorted candidates using the bitmask.
    if (tid == 0) {
        int kept = 0;
        for (int p = 0; p < KMAX && kept < MAXOUT; p++) {
            if (skey[p] <= -1e29f) break;
            if (!((suppVec[p >> 5] >> (p & 31)) & 1u)) {
                keepList[kept++] = p;
                if (kept == MAXOUT) break;
                for (int w = 0; w < KMAX / 32; w++)
                    suppVec[w] |= smask[p * (KMAX / 32) + w];
            }
        }
        while (kept < MAXOUT) keepList[kept++] = 0;
    }
    __syncthreads();

    // Outputs: boxes[100*4], labels[100] (as float), scores[100].
    if (tid < MAXOUT) {
        int p = keepList[tid];
        out[tid * 4 + 0] = bx1[p];
        out[tid * 4 + 1] = by1[p];
        out[tid * 4 + 2] = bx2[p];
        out[tid * 4 + 3] = by2[p];
        out[4 * MAXOUT + tid] = (float)labels[sidx[p]];
        out[5 * MAXOUT + tid] = skey[p];
    }
}

// ---------------------------------------------------------------- launch
extern "C" void kernel_launch(void* const* d_in, const int* in_sizes, int n_in,
                              void* d_out, int out_size, void* d_ws, size_t ws_size,
                              hipStream_t stream) {
    const float* logits  = (const float*)d_in[0];   // [N,91]
    const float* reg     = (const float*)d_in[1];   // [N,4]
    const float* ctrness = (const float*)d_in[2];   // [N,1]
    const float* anchors = (const float*)d_in[3];   // [N,4]
    const int n = in_sizes[2];                      // box_ctrness has N elems

    // workspace layout (d_ws is 256B-aligned from hipMalloc)
    float* ws_boxes  = (float*)d_ws;                // N*4
    float* ws_scores = ws_boxes + (size_t)n * 4;    // N
    int*   ws_labels = (int*)(ws_scores + n);       // N
    int*   hist      = ws_labels + n;               // NBINS
    int*   counter   = hist + NBINS;                // 1
    int*   thrbin    = counter + 1;                 // 1
    int*   cand      = thrbin + 1;                  // KMAX

    nms_init<<<1, 256, 0, stream>>>(hist, counter, thrbin);

    const int rowsPerBlock = 8;  // 256 threads = 8 wave32 rows
    fcos_decode_score<<<(n + rowsPerBlock - 1) / rowsPerBlock, 256, 0, stream>>>(
        logits, reg, ctrness, anchors, ws_boxes, ws_scores, ws_labels, n);

    score_hist<<<(n + 255) / 256, 256, 0, stream>>>(ws_scores, hist, n);
    pick_threshold<<<1, 32, 0, stream>>>(hist, thrbin);
    compact_cands<<<(n + 255) / 256, 256, 0, stream>>>(ws_scores, thrbin,
                                                       counter, cand, n);
    nms_select<<<1, 256, 0, stream>>>(ws_boxes, ws_scores, ws_labels, cand,
                                      counter, (float*)d_out);
}